// HandKineticLayer_18545668784560
// MI455X (gfx1250) — compile-verified
//
#include <hip/hip_runtime.h>
#include <stdint.h>

#ifndef __has_builtin
#define __has_builtin(x) 0
#endif

#define NPAIRS   195
#define WPB      8      // waves per block (wave32)
#define ROW_IN   63
#define ROW_OUT  258
#define LDS_ROW_F 84    // 21 landmarks x 4 floats (xyz + pad): 16B-aligned slot per landmark

// ---- compile-time pair table: j in 0..19, skip adjacent unless fingertip ----
struct PairTab { unsigned short pk[NPAIRS]; };

constexpr PairTab make_pairs() {
    PairTab t{};
    int n = 0;
    for (int j = 0; j < 20; ++j) {
        const bool tip = (j == 4) || (j == 8) || (j == 12) || (j == 16) || (j == 19);
        const int start = tip ? (j + 1) : (j + 2);
        for (int k = start; k < 21; ++k)
            t.pk[n++] = (unsigned short)(j | (k << 8));
    }
    return t;
}
__constant__ PairTab c_pairs = make_pairs();

// ---- CDNA5 async global->LDS copy (ASYNCcnt path), with portable fallback ----
#if defined(__AMDGCN__) && __has_builtin(__builtin_amdgcn_global_load_async_to_lds_b32)
#define HAVE_ASYNC_LDS 1
#else
#define HAVE_ASYNC_LDS 0
#endif

typedef __attribute__((address_space(1))) int gint_t;  // global i32
typedef __attribute__((address_space(3))) int lint_t;  // LDS i32

__device__ __forceinline__ void cp_g2lds_f32(float* lds_dst, const float* gsrc) {
#if HAVE_ASYNC_LDS
    // generic LDS pointer low 32 bits == LDS byte address; inttoptr into as(3)
    __builtin_amdgcn_global_load_async_to_lds_b32(
        (gint_t*)(uintptr_t)gsrc,
        (lint_t*)(unsigned)(uintptr_t)lds_dst,
        0, 0);
#else
    *lds_dst = gsrc[0];
#endif
}

__device__ __forceinline__ void cp_wait() {
#if HAVE_ASYNC_LDS
#if __has_builtin(__builtin_amdgcn_s_wait_asynccnt)
    __builtin_amdgcn_s_wait_asynccnt(0);
#else
    asm volatile("s_wait_asynccnt 0" ::: "memory");
#endif
#endif
}

// ---- kernel: one wave per row; private LDS slice per wave; no barriers ----
__global__ __launch_bounds__(256) void hand_kinetic_kernel(
        const float* __restrict__ in, float* __restrict__ out, int rows) {
    __shared__ float s_xyz[WPB * LDS_ROW_F];

    const int tid  = threadIdx.x;
    const int wave = tid >> 5;
    const int lane = tid & 31;
    const int row  = blockIdx.x * WPB + wave;
    if (row >= rows) return;   // uniform per wave

    const float* __restrict__ src = in  + (size_t)row * ROW_IN;
    float*       __restrict__ dst = out + (size_t)row * ROW_OUT;
    float* lrow = &s_xyz[wave * LDS_ROW_F];

    // Stage 63 floats into LDS, deinterleaved: element c=3k+r -> slot 4k+r
    // (each landmark's (x,y) becomes one 8B-aligned ds_load_b64 later).
    const int c0 = lane;                    // 0..31
    const int k0 = c0 / 3;
    const int s0 = k0 * 4 + (c0 - 3 * k0);
    cp_g2lds_f32(lrow + s0, src + c0);

    const int c1 = lane + 32;               // 32..63
    const int k1 = c1 / 3;
    const int s1 = k1 * 4 + (c1 - 3 * k1);
    if (c1 < ROW_IN)
        cp_g2lds_f32(lrow + s1, src + c1);

    cp_wait();                              // s_wait_asynccnt 0 (own wave's copies)

    // Pass-through copy of the 63 inputs (coalesced non-temporal stores:
    // 1.03 GB output stream >> 192 MB L2, don't pollute it).
    __builtin_nontemporal_store(lrow[s0], dst + c0);
    if (c1 < ROW_IN)
        __builtin_nontemporal_store(lrow[s1], dst + c1);

    // 195 pairwise 2-D distances: 7 x 32 lanes, 2 x ds_load_b64 per pair.
    #pragma unroll
    for (int t = 0; t < 7; ++t) {
        const int p = t * 32 + lane;
        if (p < NPAIRS) {
            const unsigned pk = c_pairs.pk[p];
            const int i = (int)(pk & 0xffu);
            const int j = (int)(pk >> 8);
            const float2 a = *(const float2*)&lrow[i * 4];  // aligned b64 LDS read
            const float2 b = *(const float2*)&lrow[j * 4];
            const float dx = a.x - b.x;
            const float dy = a.y - b.y;
            const float d  = __builtin_amdgcn_sqrtf(__builtin_fmaf(dx, dx, dy * dy));
            __builtin_nontemporal_store(d, dst + ROW_IN + p);
        }
    }
}

extern "C" void kernel_launch(void* const* d_in, const int* in_sizes, int n_in,
                              void* d_out, int out_size, void* d_ws, size_t ws_size,
                              hipStream_t stream) {
    const float* in  = (const float*)d_in[0];
    float*       out = (float*)d_out;
    const int rows   = in_sizes[0] / ROW_IN;          // B = 1,000,000
    const int blocks = (rows + WPB - 1) / WPB;
    hipLaunchKernelGGL(hand_kinetic_kernel, dim3(blocks), dim3(256), 0, stream,
                       in, out, rows);
}